// EuclideanGraphBuilder_12945031430844
// MI455X (gfx1250) — compile-verified
//
#include <hip/hip_runtime.h>
#include <hip/hip_bf16.h>

typedef float v2f __attribute__((ext_vector_type(2)));
typedef float v8f __attribute__((ext_vector_type(8)));

#define FDIM 6
#define THRESH 1e-4f

// ---------------------------------------------------------------------------
// zero the row-sum workspace
// ---------------------------------------------------------------------------
__global__ void egb_zero_kernel(float* __restrict__ S, int n) {
    int i = blockIdx.x * blockDim.x + threadIdx.x;
    if (i < n) S[i] = 0.0f;
}

// ---------------------------------------------------------------------------
// helpers: build the per-lane WMMA A/B fragment for a 16-row strip of xyz
// (K=3 padded to 4). Lanes 0-15 hold (x,y) = K0,K1 of row base+lane;
// lanes 16-31 hold (z,0) = K2,K3 of row base+(lane-16).
// ---------------------------------------------------------------------------
__device__ inline void load_strip(const float* __restrict__ x, int base, int lo, int hi,
                                  v2f& frag, float& sq) {
    int r = base + lo;
    float xr = x[r * FDIM + 0];
    float yr = x[r * FDIM + 1];
    float zr = x[r * FDIM + 2];
    sq = xr * xr + yr * yr + zr * zr;
    frag.x = hi ? zr : xr;
    frag.y = hi ? 0.0f : yr;
}

// ---------------------------------------------------------------------------
// Pass 1: S[i] = sum_j exp(exp(-2 * d2(i,j)))
// one block (8 waves) per 16-row strip; waves split the column tiles
// ---------------------------------------------------------------------------
__global__ void egb_rowsum_kernel(const float* __restrict__ x,
                                  float* __restrict__ S, int n) {
    const int lane = threadIdx.x & 31;
    const int wave = threadIdx.x >> 5;
    const int lo = lane & 15;
    const int hi = lane >> 4;
    const int I = blockIdx.x * 16;

    v2f a; float sqi;
    load_strip(x, I, lo, hi, a, sqi);

    // broadcast |x_row|^2 to the (v, hi) slots of the C/D layout
    float sqI[8];
#pragma unroll
    for (int v = 0; v < 8; ++v) sqI[v] = __shfl(sqi, v + 8 * hi, 32);

    float acc[8];
#pragma unroll
    for (int v = 0; v < 8; ++v) acc[v] = 0.0f;

    const int nJt = n >> 4;
    for (int jt = wave; jt < nJt; jt += 8) {
        v2f b; float sqj;
        load_strip(x, jt * 16, lo, hi, b, sqj);

        v8f d = {};
        d = __builtin_amdgcn_wmma_f32_16x16x4_f32(
            /*neg_a=*/false, a, /*neg_b=*/false, b,
            /*c_mod=*/(short)0, d, /*reuse_a=*/false, /*reuse_b=*/false);

#pragma unroll
        for (int v = 0; v < 8; ++v) {
            float d2 = sqI[v] + sqj - 2.0f * d[v];
            d2 = fmaxf(d2, 0.0f);
            acc[v] += __expf(__expf(-2.0f * d2));
        }
    }

    // reduce over the 16 column-lanes of each half-wave, then atomically add
#pragma unroll
    for (int v = 0; v < 8; ++v) {
        float s = acc[v];
#pragma unroll
        for (int off = 1; off < 16; off <<= 1) s += __shfl_xor(s, off, 32);
        if (lo == 0) atomicAdd(&S[I + v + 8 * hi], s);
    }
}

// ---------------------------------------------------------------------------
// Pass 2: out[i,j] = masked( exp(exp(-2*d2)) / S[i] )
// grid (n/16, n/16/8); 8 waves per block, one 16x16 tile per wave
// ---------------------------------------------------------------------------
__global__ void egb_write_kernel(const float* __restrict__ x,
                                 const int* __restrict__ batch,
                                 const float* __restrict__ S,
                                 float* __restrict__ out, int n) {
    const int lane = threadIdx.x & 31;
    const int wave = threadIdx.x >> 5;
    const int lo = lane & 15;
    const int hi = lane >> 4;
    const int I = blockIdx.x * 16;
    const int J = (blockIdx.y * 8 + wave) * 16;

    v2f a; float sqi;
    load_strip(x, I, lo, hi, a, sqi);
    float s_own = S[I + lo];
    int   b_own = batch[I + lo];

    float sqI[8], Sinv[8]; int bI[8];
#pragma unroll
    for (int v = 0; v < 8; ++v) {
        int src = v + 8 * hi;
        sqI[v]  = __shfl(sqi, src, 32);
        Sinv[v] = 1.0f / __shfl(s_own, src, 32);
        bI[v]   = __shfl(b_own, src, 32);
    }

    v2f b; float sqj;
    load_strip(x, J, lo, hi, b, sqj);
    int bj = batch[J + lo];

    v8f d = {};
    d = __builtin_amdgcn_wmma_f32_16x16x4_f32(
        false, a, false, b, (short)0, d, false, false);

#pragma unroll
    for (int v = 0; v < 8; ++v) {
        float d2 = sqI[v] + sqj - 2.0f * d[v];
        d2 = fmaxf(d2, 0.0f);
        float w = __expf(__expf(-2.0f * d2)) * Sinv[v];
        bool ok = (w > THRESH) && (bI[v] == bj);
        size_t row = (size_t)(I + v + 8 * hi);
        out[row * (size_t)n + (size_t)(J + lo)] = ok ? w : 0.0f;
    }
}

// ---------------------------------------------------------------------------
extern "C" void kernel_launch(void* const* d_in, const int* in_sizes, int n_in,
                              void* d_out, int out_size, void* d_ws, size_t ws_size,
                              hipStream_t stream) {
    const float* x     = (const float*)d_in[0];
    const int*   batch = (const int*)d_in[1];
    float*       out   = (float*)d_out;
    float*       S     = (float*)d_ws;

    const int n = in_sizes[0] / FDIM;   // 8192
    const int nTiles = n / 16;          // 512

    egb_zero_kernel<<<(n + 255) / 256, 256, 0, stream>>>(S, n);
    egb_rowsum_kernel<<<nTiles, 256, 0, stream>>>(x, S, n);
    egb_write_kernel<<<dim3(nTiles, nTiles / 8), 256, 0, stream>>>(x, batch, S, out, n);
}